// ALE_1176821039620
// MI455X (gfx1250) — compile-verified
//
#include <hip/hip_runtime.h>

#define BLOCK 256

// --- Diagnostic probe (never launched): placed FIRST so the disasm snippet
// shows exactly how atomicAdd(float*) lowers on gfx1250 (native
// global_atomic_add_f32 vs. a cmpswap retry loop).
__global__ void gp_atomic_probe(float* p, float v) {
  atomicAdd(p, v);
}

// One propagation step: nxt[dst[e]] += cur[src[e]] * ep[e], 4 edges per thread.
__global__ __launch_bounds__(BLOCK)
void gp_edge_kernel(const int* __restrict__ src, const int* __restrict__ dst,
                    const float* __restrict__ ep, const float* __restrict__ cur,
                    float* __restrict__ nxt, int E) {
  int t = blockIdx.x * BLOCK + threadIdx.x;
  int base = t * 4;
  if (base >= E) return;

  if (base + 4096 + 3 < E) {
    // Stream-ahead prefetch of the edge arrays (gfx1250 global_prefetch_b8,
    // SCOPE_SYS -> pulls the line into GL2, where the whole 77MB edge stream
    // stays resident across all 4 passes).
    __builtin_prefetch(&src[base + 4096], 0, 0);
    __builtin_prefetch(&dst[base + 4096], 0, 0);
    __builtin_prefetch(&ep [base + 4096], 0, 0);
  }

  if (base + 3 < E && (E & 3) == 0) {
    // Fast path: B128 loads of the three edge streams.
    int4   s = reinterpret_cast<const int4*  >(src)[t];
    int4   d = reinterpret_cast<const int4*  >(dst)[t];
    float4 p = reinterpret_cast<const float4*>(ep )[t];
    atomicAdd(&nxt[d.x], cur[s.x] * p.x);
    atomicAdd(&nxt[d.y], cur[s.y] * p.y);
    atomicAdd(&nxt[d.z], cur[s.z] * p.z);
    atomicAdd(&nxt[d.w], cur[s.w] * p.w);
  } else {
    for (int e = base; e < E && e < base + 4; ++e) {
      atomicAdd(&nxt[dst[e]], cur[src[e]] * ep[e]);
    }
  }
}

// out[i] = w[0]*x[i]; zero both ping-pong accumulators.
__global__ __launch_bounds__(BLOCK)
void gp_init_kernel(const float* __restrict__ x, const float* __restrict__ w,
                    float* __restrict__ out, float* __restrict__ A,
                    float* __restrict__ B, int n) {
  int i = blockIdx.x * BLOCK + threadIdx.x;
  if (i < n) {
    out[i] = w[0] * x[i];
    A[i] = 0.0f;
    B[i] = 0.0f;
  }
}

// out[i] += w[k] * acc[i]; optionally zero the other buffer for the next step.
__global__ __launch_bounds__(BLOCK)
void gp_node_kernel(const float* __restrict__ acc, const float* __restrict__ w,
                    int k, float* __restrict__ out, float* __restrict__ zero_buf,
                    int n) {
  int i = blockIdx.x * BLOCK + threadIdx.x;
  if (i < n) {
    out[i] += w[k] * acc[i];
    if (zero_buf) zero_buf[i] = 0.0f;
  }
}

extern "C" void kernel_launch(void* const* d_in, const int* in_sizes, int n_in,
                              void* d_out, int out_size, void* d_ws, size_t ws_size,
                              hipStream_t stream) {
  const float* x  = (const float*)d_in[0];
  const int*   ei = (const int*  )d_in[1];  // (2, E): row0 = src, row1 = dst
  const float* ep = (const float*)d_in[2];
  const float* w  = (const float*)d_in[3];  // (5,)

  const int N = in_sizes[0];                // 100000 (x is (N,1))
  const int E = in_sizes[2];                // 6400000

  const int* src = ei;
  const int* dst = ei + E;

  float* out = (float*)d_out;
  float* A   = (float*)d_ws;                // ping
  float* B   = A + N;                       // pong

  const int nb_n = (N + BLOCK - 1) / BLOCK;
  const int e4   = (E + 3) / 4;
  const int nb_e = (e4 + BLOCK - 1) / BLOCK;

  // result = w0*x; A = B = 0
  gp_init_kernel<<<nb_n, BLOCK, 0, stream>>>(x, w, out, A, B, N);

  // k=1: gather x  -> scatter A ; out += w1*A ; zero B
  gp_edge_kernel<<<nb_e, BLOCK, 0, stream>>>(src, dst, ep, x, A, E);
  gp_node_kernel<<<nb_n, BLOCK, 0, stream>>>(A, w, 1, out, B, N);

  // k=2: gather A  -> scatter B ; out += w2*B ; zero A
  gp_edge_kernel<<<nb_e, BLOCK, 0, stream>>>(src, dst, ep, A, B, E);
  gp_node_kernel<<<nb_n, BLOCK, 0, stream>>>(B, w, 2, out, A, N);

  // k=3: gather B  -> scatter A ; out += w3*A ; zero B
  gp_edge_kernel<<<nb_e, BLOCK, 0, stream>>>(src, dst, ep, B, A, E);
  gp_node_kernel<<<nb_n, BLOCK, 0, stream>>>(A, w, 3, out, B, N);

  // k=4: gather A  -> scatter B ; out += w4*B
  gp_edge_kernel<<<nb_e, BLOCK, 0, stream>>>(src, dst, ep, A, B, E);
  gp_node_kernel<<<nb_n, BLOCK, 0, stream>>>(B, w, 4, out, nullptr, N);
}